// GATv2Block_60687887893139
// MI455X (gfx1250) — compile-verified
//
#include <hip/hip_runtime.h>

#define N_NODES   100000
#define IN_C      128
#define HEADS     4
#define CPH       16
#define HC        64          // HEADS*CPH
#define E_EDGES   1600000
#define ET        (E_EDGES + N_NODES)   // with self loops
#define NEG_SLOPE 0.2f

typedef __attribute__((ext_vector_type(2))) float v2f;
typedef __attribute__((ext_vector_type(8))) float v8f;

// ---------------- monotone float<->int key (for atomicMax on float) --------
__device__ __forceinline__ int f2key(float f) {
    int i = __float_as_int(f);
    return i ^ ((i >> 31) & 0x7fffffff);
}
__device__ __forceinline__ float key2f(int k) {
    return __int_as_float(k ^ ((k >> 31) & 0x7fffffff));
}

// ---------------- kernel 0: init workspace ---------------------------------
__global__ __launch_bounds__(256) void k_init(float* __restrict__ out_acc,
                                              float* __restrict__ denom,
                                              int*   __restrict__ emaxk,
                                              float* __restrict__ stats) {
    size_t i = (size_t)blockIdx.x * 256 + threadIdx.x;
    if (i < (size_t)N_NODES * HC) out_acc[i] = 0.0f;
    if (i < (size_t)N_NODES * HEADS) { denom[i] = 0.0f; emaxk[i] = (int)0x80000000; }
    if (i < 2 * HC) stats[i] = 0.0f;
}

// ---------------- kernel 1: x_l = x@W_l + b_l ; x_r = x@W_r + b_r (WMMA) ---
// block = 256 threads (8 waves); 64 rows of x per block staged in LDS.
// waves 0..3 -> (W_l, x_l) col tiles 0..3 ; waves 4..7 -> (W_r, x_r).
// All 32 B fragments (K=128 in steps of 4) are preloaded into registers once
// per wave and reused across the 4 row tiles; inner loop is ds_load + wmma.
__global__ __launch_bounds__(256) void k_linear_wmma(
        const float* __restrict__ x,
        const float* __restrict__ Wl, const float* __restrict__ bl,
        const float* __restrict__ Wr, const float* __restrict__ br,
        float* __restrict__ xl, float* __restrict__ xr) {
    __shared__ float xs[64 * IN_C];          // 32 KB
    const int tid  = threadIdx.x;
    const int row0 = blockIdx.x * 64;

    // cooperative, coalesced load of the 64x128 tile (zero-padded at the tail)
    const float4* xg  = (const float4*)x;
    float4*       xs4 = (float4*)xs;
    for (int i = tid; i < 64 * (IN_C / 4); i += 256) {
        int r = i / (IN_C / 4);
        float4 v = make_float4(0.f, 0.f, 0.f, 0.f);
        if (row0 + r < N_NODES)
            v = xg[(size_t)(row0 + r) * (IN_C / 4) + (i % (IN_C / 4))];
        xs4[i] = v;
    }
    __syncthreads();

    const int wave = tid >> 5;          // 0..7
    const int lane = tid & 31;
    const int m8   = lane >> 4;         // 0 or 1 (half-wave)
    const int l16  = lane & 15;
    const bool right   = wave >= 4;
    const float* W     = right ? Wr : Wl;
    const float* bias  = right ? br : bl;
    float*       out   = right ? xr : xl;
    const int    col   = (wave & 3) * 16 + l16;
    const float  bcol  = bias[col];

    // Preload all B fragments: 4x16 f32, lanes 0-15 K=kk..kk+1, lanes 16-31
    // K=kk+2..kk+3 (W row-major [128 x 64]). 32 x v2f = 64 VGPRs.
    v2f bfrag[IN_C / 4];
    #pragma unroll
    for (int k4 = 0; k4 < IN_C / 4; ++k4) {
        const int ak = k4 * 4 + 2 * m8;
        bfrag[k4].x = W[(size_t)(ak)     * HC + col];
        bfrag[k4].y = W[(size_t)(ak + 1) * HC + col];
    }

    for (int rt = 0; rt < 4; ++rt) {
        v8f acc;
        #pragma unroll
        for (int r = 0; r < 8; ++r) acc[r] = bcol;   // C initialized with bias

        #pragma unroll
        for (int k4 = 0; k4 < IN_C / 4; ++k4) {
            // A frag 16x4 f32 from LDS: lanes 0-15 K=kk..kk+1, 16-31 K=kk+2..kk+3
            const int ar = rt * 16 + l16;
            const int ak = k4 * 4 + 2 * m8;
            v2f a;
            a.x = xs[ar * IN_C + ak];
            a.y = xs[ar * IN_C + ak + 1];
            acc = __builtin_amdgcn_wmma_f32_16x16x4_f32(
                      false, a, false, bfrag[k4], (short)0, acc, false, false);
        }
        // D layout: VGPR r -> row (rt*16 + r + 8*m8), col
        #pragma unroll
        for (int r = 0; r < 8; ++r) {
            int row = row0 + rt * 16 + r + 8 * m8;
            if (row < N_NODES) out[(size_t)row * HC + col] = acc[r];
        }
    }
}

// ---------------- kernel 2: per-edge attention scores + segment max --------
__global__ __launch_bounds__(256) void k_edge_scores(
        const int* __restrict__ ei,
        const float* __restrict__ xl, const float* __restrict__ xr,
        const float* __restrict__ att,
        float* __restrict__ e_out, int* __restrict__ emaxk) {
    long long t = (long long)blockIdx.x * 256 + threadIdx.x;
    if (t >= ET) return;
    int src, dst;
    if (t < E_EDGES) { src = ei[t]; dst = ei[E_EDGES + t]; }
    else             { src = dst = (int)(t - E_EDGES); }

    const float4* a4   = (const float4*)xl + (size_t)src * (HC / 4);
    const float4* b4   = (const float4*)xr + (size_t)dst * (HC / 4);
    const float4* att4 = (const float4*)att;

    #pragma unroll
    for (int h = 0; h < HEADS; ++h) {
        float s = 0.0f;
        #pragma unroll
        for (int q = 0; q < 4; ++q) {
            float4 a = a4[h * 4 + q];
            float4 b = b4[h * 4 + q];
            float4 w = att4[h * 4 + q];
            float mx = a.x + b.x; mx = mx > 0.f ? mx : NEG_SLOPE * mx;
            float my = a.y + b.y; my = my > 0.f ? my : NEG_SLOPE * my;
            float mz = a.z + b.z; mz = mz > 0.f ? mz : NEG_SLOPE * mz;
            float mw = a.w + b.w; mw = mw > 0.f ? mw : NEG_SLOPE * mw;
            s += mx * w.x + my * w.y + mz * w.z + mw * w.w;
        }
        e_out[(size_t)t * HEADS + h] = s;
        atomicMax(&emaxk[(size_t)dst * HEADS + h], f2key(s));
    }
}

// ---------------- kernel 3: exp-weights, denom and UNNORMALIZED aggregation
__global__ __launch_bounds__(256) void k_edge_accum(
        const int* __restrict__ ei,
        const float* __restrict__ xl,
        const float* __restrict__ e_in, const int* __restrict__ emaxk,
        float* __restrict__ denom, float* __restrict__ out_acc) {
    long long t = (long long)blockIdx.x * 256 + threadIdx.x;
    if (t >= ET) return;
    int src, dst;
    if (t < E_EDGES) { src = ei[t]; dst = ei[E_EDGES + t]; }
    else             { src = dst = (int)(t - E_EDGES); }

    float w[HEADS];
    #pragma unroll
    for (int h = 0; h < HEADS; ++h) {
        float mx = key2f(emaxk[(size_t)dst * HEADS + h]);
        w[h] = __expf(e_in[(size_t)t * HEADS + h] - mx);
        atomicAdd(&denom[(size_t)dst * HEADS + h], w[h]);
    }
    const float* a = xl + (size_t)src * HC;
    float*       o = out_acc + (size_t)dst * HC;
    #pragma unroll
    for (int k = 0; k < HC; ++k)
        atomicAdd(&o[k], w[k >> 4] * a[k]);
}

// ---------------- kernel 4: divide by denom, add bias, channel stat reduce -
__global__ __launch_bounds__(256) void k_node_stats(
        const float* __restrict__ out_acc, const float* __restrict__ denom,
        const float* __restrict__ bias_out,
        float* __restrict__ out_tmp, float* __restrict__ stats) {
    __shared__ float s1[HC];
    __shared__ float s2[HC];
    const int tid = threadIdx.x;
    if (tid < HC) { s1[tid] = 0.0f; s2[tid] = 0.0f; }
    __syncthreads();

    size_t i  = (size_t)blockIdx.x * 256 + tid;
    const int ch = tid & (HC - 1);          // 256 % 64 == 0
    if (i < (size_t)N_NODES * HC) {
        size_t node = i / HC;
        float o = out_acc[i] / (denom[node * HEADS + (ch >> 4)] + 1e-16f)
                  + bias_out[ch];
        out_tmp[i] = o;
        atomicAdd(&s1[ch], o);              // LDS float atomics (ds_add_f32)
        atomicAdd(&s2[ch], o * o);
    }
    __syncthreads();
    if (tid < HC) {
        atomicAdd(&stats[tid],      s1[tid]);
        atomicAdd(&stats[HC + tid], s2[tid]);
    }
}

// ---------------- kernel 5: GraphNorm + ReLU (in place on d_out) -----------
__global__ __launch_bounds__(256) void k_node_norm(
        float* __restrict__ out, const float* __restrict__ stats,
        const float* __restrict__ gnw, const float* __restrict__ gnb,
        const float* __restrict__ gms) {
    size_t i = (size_t)blockIdx.x * 256 + threadIdx.x;
    if (i >= (size_t)N_NODES * HC) return;
    const int ch = (int)(i & (HC - 1));
    const float invN = 1.0f / (float)N_NODES;
    float mean = stats[ch] * invN;
    float a    = gms[ch] * mean;
    // E[(o-a)^2] = E[o^2] - 2 a mean + a^2
    float var  = stats[HC + ch] * invN - 2.0f * a * mean + a * a;
    float o    = (out[i] - a) * rsqrtf(var + 1e-5f) * gnw[ch] + gnb[ch];
    out[i] = o > 0.0f ? o : 0.0f;
}

// ---------------------------------------------------------------------------
extern "C" void kernel_launch(void* const* d_in, const int* in_sizes, int n_in,
                              void* d_out, int out_size, void* d_ws, size_t ws_size,
                              hipStream_t stream) {
    const float* x        = (const float*)d_in[0];
    const int*   ei       = (const int*)  d_in[1];   // [2,E]: src row then dst row
    const float* Wl       = (const float*)d_in[2];
    const float* bl       = (const float*)d_in[3];
    const float* Wr       = (const float*)d_in[4];
    const float* br       = (const float*)d_in[5];
    const float* att      = (const float*)d_in[6];
    const float* bias_out = (const float*)d_in[7];
    const float* gnw      = (const float*)d_in[8];
    const float* gnb      = (const float*)d_in[9];
    const float* gms      = (const float*)d_in[10];
    float*       out      = (float*)d_out;

    // workspace layout (floats): ~107 MB total
    float* ws      = (float*)d_ws;
    float* xl      = ws;                                  // N*64
    float* xr      = xl      + (size_t)N_NODES * HC;      // N*64
    float* e       = xr      + (size_t)N_NODES * HC;      // ET*4
    float* out_acc = e       + (size_t)ET * HEADS;        // N*64
    float* denom   = out_acc + (size_t)N_NODES * HC;      // N*4
    int*   emaxk   = (int*)(denom + (size_t)N_NODES * HEADS); // N*4
    float* stats   = (float*)(emaxk + (size_t)N_NODES * HEADS); // 128

    const int elemBlocks = (int)(((size_t)N_NODES * HC + 255) / 256);   // 25000
    const int gemmBlocks = (N_NODES + 63) / 64;                         // 1563
    const int edgeBlocks = (int)(((long long)ET + 255) / 256);          // 6641

    k_init<<<elemBlocks, 256, 0, stream>>>(out_acc, denom, emaxk, stats);
    k_linear_wmma<<<gemmBlocks, 256, 0, stream>>>(x, Wl, bl, Wr, br, xl, xr);
    k_edge_scores<<<edgeBlocks, 256, 0, stream>>>(ei, xl, xr, att, e, emaxk);
    k_edge_accum<<<edgeBlocks, 256, 0, stream>>>(ei, xl, e, emaxk, denom, out_acc);
    k_node_stats<<<elemBlocks, 256, 0, stream>>>(out_acc, denom, bias_out, out, stats);
    k_node_norm<<<elemBlocks, 256, 0, stream>>>(out, stats, gnw, gnb, gms);
}